// RQspline_7464653161051
// MI455X (gfx1250) — compile-verified
//
#include <hip/hip_runtime.h>
#include <stdint.h>

// ---------------- problem constants (from the reference) ----------------
#define NDIMS   512
#define KNOTS   32
#define NDATA_N 65536

// ---------------- tiling ----------------
#define PAD            33                     // knot-table row stride (bank-conflict padding)
#define DCHUNK         128                    // dims per block (LDS chunk)
#define TBL_DIM_FLOATS (DCHUNK * PAD)         // 4224 floats per table per chunk
#define TBL_ALL_FLOATS (3 * TBL_DIM_FLOATS)   // 12672 floats  (50688 B LDS)
#define WS_TBL_STRIDE  (NDIMS * PAD)          // 16896 floats per full table in workspace
#define CHUNK16        (TBL_DIM_FLOATS / 4)   // 1056 16-byte packets per table chunk
#define BLOCK_THREADS  256
#define ROWS_PER_PASS  8                      // 256 threads / (128 dims / 4-wide)
#define PASSES         32                     // rows per block = 256
#define ROWS_PER_BLOCK (ROWS_PER_PASS * PASSES)

typedef float v4f __attribute__((ext_vector_type(4)));
typedef int   v4i __attribute__((ext_vector_type(4)));
typedef __attribute__((address_space(1))) v4i gas_v4i;   // global int4
typedef __attribute__((address_space(3))) v4i las_v4i;   // LDS int4

// =======================================================================
// Kernel 1: build padded knot tables  ws = [xx | yy | delta], each
// [NDIMS][PAD] f32 (entries 0..31 valid).  Tiny: 512 threads total.
// =======================================================================
__global__ __launch_bounds__(512) void rq_build_tables(const float* __restrict__ params,
                                                       float* __restrict__ ws) {
  const int d = blockIdx.x * blockDim.x + threadIdx.x;
  if (d >= NDIMS) return;
  float* xx = ws + 0 * WS_TBL_STRIDE + d * PAD;
  float* yy = ws + 1 * WS_TBL_STRIDE + d * PAD;
  float* dd = ws + 2 * WS_TBL_STRIDE + d * PAD;

  const float* logdx = params + 2 * NDIMS + d * (KNOTS - 1);
  const float* logdy = logdx + NDIMS * (KNOTS - 1);
  const float* logdv = params + 2 * NDIMS + 2 * NDIMS * (KNOTS - 1) + d * KNOTS;

  float xv = params[d];
  float yv = params[NDIMS + d];
  xx[0] = xv;
  yy[0] = yv;
#pragma unroll 1
  for (int j = 0; j < KNOTS - 1; ++j) {
    xv += expf(logdx[j]); xx[j + 1] = xv;   // cumsum(exp(logdx)) like the reference
    yv += expf(logdy[j]); yy[j + 1] = yv;
  }
#pragma unroll 1
  for (int j = 0; j < KNOTS; ++j) dd[j] = expf(logdv[j]);
}

// =======================================================================
// Kernel 2: streaming RQ-spline evaluation.
//   grid = (NDATA/256 rows, 4 dim-chunks), block = 256 threads.
//   Tables for this block's 128 dims staged into LDS via the gfx1250
//   async global->LDS path (ASYNCcnt), then each thread processes
//   4 consecutive dims x 32 row-passes with NT b128 loads/stores.
// =======================================================================
__global__ __launch_bounds__(BLOCK_THREADS) void rq_spline_eval(const float* __restrict__ x,
                                                                float* __restrict__ ws,
                                                                float* __restrict__ out) {
  __shared__ __align__(16) float lds[TBL_ALL_FLOATS];

  const int tid   = threadIdx.x;
  const int chunk = blockIdx.y;         // which 128-dim slab
  const int dim0  = chunk * DCHUNK;

  // ---- stage 3 x 16896 B of tables into LDS ----
#if __has_builtin(__builtin_amdgcn_global_load_async_to_lds_b128)
  {
    // two-step casts: generic reinterpret, then addrspace cast
    gas_v4i* gbase = (gas_v4i*)(v4i*)ws;
    las_v4i* lbase = (las_v4i*)(v4i*)lds;
    const int gofs4 = (dim0 * PAD) >> 2;            // chunk offset in int4 units
    for (int i = tid; i < 3 * CHUNK16; i += BLOCK_THREADS) {
      const int t = i / CHUNK16;                    // table 0..2
      const int r = i - t * CHUNK16;                // 16B packet within table chunk
      gas_v4i* src = gbase + (t * (WS_TBL_STRIDE >> 2)) + gofs4 + r;
      las_v4i* dst = lbase + (t * (TBL_DIM_FLOATS >> 2)) + r;
      __builtin_amdgcn_global_load_async_to_lds_b128(src, dst, 0, 0);
    }
    asm volatile("s_wait_asynccnt 0" ::: "memory");
  }
#else
  for (int i = tid; i < 3 * CHUNK16; i += BLOCK_THREADS) {
    const int t = i / CHUNK16;
    const int r = i - t * CHUNK16;
    const v4f* src = (const v4f*)(ws + (size_t)t * WS_TBL_STRIDE + (size_t)dim0 * PAD) + r;
    *(v4f*)&lds[(size_t)t * TBL_DIM_FLOATS + (size_t)r * 4] = *src;
  }
#endif
  __syncthreads();

  const float* xxl = lds;
  const float* yyl = lds + TBL_DIM_FLOATS;
  const float* ddl = lds + 2 * TBL_DIM_FLOATS;

  const int lane = tid & 31;
  const int rloc = tid >> 5;
  const int dloc = lane * 4;                       // local dim base (0..124)
  const size_t row_base = (size_t)blockIdx.x * ROWS_PER_BLOCK;

  float* yout = out;
  float* lout = out + (size_t)NDATA_N * NDIMS;

#pragma unroll 1
  for (int it = 0; it < PASSES; ++it) {
    const size_t row  = row_base + (size_t)it * ROWS_PER_PASS + rloc;
    const size_t base = row * NDIMS + dim0 + dloc;

    const v4f xv = __builtin_nontemporal_load((const v4f*)(x + base));
    v4f y4, l4;

#pragma unroll
    for (int e = 0; e < 4; ++e) {
      const float xs   = xv[e];
      const int   toff = (dloc + e) * PAD;
      const float* t   = xxl + toff;

      // branchless lower_bound over 32 knots: p = min(#knots < xs, 31)
      int p = 0;
      p += (t[p + 15] < xs) ? 16 : 0;
      p += (t[p + 7]  < xs) ? 8  : 0;
      p += (t[p + 3]  < xs) ? 4  : 0;
      p += (t[p + 1]  < xs) ? 2  : 0;
      p += (t[p]      < xs) ? 1  : 0;
      const int ic = (p < 1) ? 1 : p;              // clamp(index,1,31)

      const float x_lo = t[ic - 1],          x_hi = t[ic];
      const float y_lo = yyl[toff + ic - 1], y_hi = yyl[toff + ic];
      const float d_lo = ddl[toff + ic - 1], d_hi = ddl[toff + ic];

      const bool sel0 = (xs <= x_lo);              // == (searchsorted index == 0)
      const bool selN = (xs >  x_hi);              // == (searchsorted index == 32)

      const float w = x_hi - x_lo;
      float invw = __builtin_amdgcn_rcpf(w);
      invw = invw * (2.0f - w * invw);             // 1 Newton step

      const float xc   = fminf(fmaxf(xs, x_lo), x_hi);
      const float xi   = (xc - x_lo) * invw;
      const float s    = (y_hi - y_lo) * invw;
      const float omxi = 1.0f - xi;
      const float xi1  = xi * omxi;
      const float xi2  = xi * xi;

      const float dsum = d_hi + d_lo - 2.0f * s;
      const float den  = fmaf(dsum, xi1, s);
      float invden = __builtin_amdgcn_rcpf(den);
      invden = invden * (2.0f - den * invden);

      const float num_y = fmaf(s, xi2, d_lo * xi1);
      const float y_mid = fmaf(y_hi - y_lo, num_y * invden, y_lo);

      // 2*log(s) + log(num) - 2*log(den)  ==  log(s^2 * num / den^2)
      const float num_l = fmaf(d_hi, xi2, fmaf(2.0f * s, xi1, d_lo * omxi * omxi));
      const float larg  = sel0 ? d_lo
                        : selN ? d_hi
                               : (s * s) * num_l * (invden * invden);
      const float lval  = __logf(larg);

      const float yv_ = sel0 ? fmaf(xs - x_lo, d_lo, y_lo)     // lower linear tail
                      : selN ? fmaf(xs - x_hi, d_hi, y_hi)     // upper linear tail
                             : y_mid;
      y4[e] = yv_;
      l4[e] = lval;
    }

    __builtin_nontemporal_store(y4, (v4f*)(yout + base));
    __builtin_nontemporal_store(l4, (v4f*)(lout + base));
  }
}

// =======================================================================
extern "C" void kernel_launch(void* const* d_in, const int* in_sizes, int n_in,
                              void* d_out, int out_size, void* d_ws, size_t ws_size,
                              hipStream_t stream) {
  const float* x      = (const float*)d_in[0];
  const float* params = (const float*)d_in[1];
  float*       out    = (float*)d_out;
  float*       ws     = (float*)d_ws;     // needs 3*512*33*4 = 202752 B

  hipLaunchKernelGGL(rq_build_tables, dim3(1), dim3(512), 0, stream, params, ws);

  dim3 grid(NDATA_N / ROWS_PER_BLOCK, NDIMS / DCHUNK);   // (256, 4)
  hipLaunchKernelGGL(rq_spline_eval, grid, dim3(BLOCK_THREADS), 0, stream, x, ws, out);
}